// S6Layer_50818053046279
// MI455X (gfx1250) — compile-verified
//
#include <hip/hip_runtime.h>
#include <hip/hip_bf16.h>
#include <math.h>

typedef __attribute__((ext_vector_type(2))) float v2f;
typedef __attribute__((ext_vector_type(8))) float v8f;

#define LSEQ 4096
#define DDIM 768
#define NS   16
#define RR   48
#define BCDT_W 80        // [B(16) | C(16) | dt_low(48)]
#define NCH  64
#define CHLEN (LSEQ / NCH)   // 64

__device__ __forceinline__ float softplus_f(float v) {
    return v > 20.0f ? v : log1pf(expf(v));
}
__device__ __forceinline__ float soft_clamp_f(float v) {
    const float lo = 1e-4f, hi = 10.0f;
    const float c  = 0.5f * (hi + lo);
    const float hr = 0.5f * (hi - lo);
    return c + hr * tanhf((v - c) / hr);
}

// ---------------- Kernel 1: bcdt = x @ [W_bc | W_dt]  (f32 WMMA) ----------------
// One wave per 16x16 output tile. 5 N-tiles (2 from W_bc, 3 from W_dt).
__global__ void __launch_bounds__(32)
gemm_bcdt(const float* __restrict__ x, const float* __restrict__ Wbc,
          const float* __restrict__ Wdt, float* __restrict__ bcdt) {
    const int nt = blockIdx.x % 5;
    const int mt = blockIdx.x / 5;
    const int laneM  = threadIdx.x & 15;
    const int laneHi = threadIdx.x >> 4;
    const float* W; int wstride, col0;
    if (nt < 2) { W = Wbc; wstride = 2 * NS; col0 = nt * 16; }
    else        { W = Wdt; wstride = RR;     col0 = (nt - 2) * 16; }
    const int m0 = mt * 16;
    v8f acc = {};
    for (int k0 = 0; k0 < DDIM; k0 += 4) {
        const int ka = k0 + 2 * laneHi;           // ISA 16x4 f32 A layout
        v2f a, b;
        a.x = x[(m0 + laneM) * DDIM + ka];
        a.y = x[(m0 + laneM) * DDIM + ka + 1];
        b.x = W[(ka    ) * wstride + col0 + laneM];
        b.y = W[(ka + 1) * wstride + col0 + laneM];
        acc = __builtin_amdgcn_wmma_f32_16x16x4_f32(false, a, false, b,
                                                    (short)0, acc, false, false);
    }
#pragma unroll
    for (int v = 0; v < 8; ++v) {
        const int t = m0 + v + 8 * laneHi;        // C/D: M = vgpr + 8*laneHi, N = laneM
        bcdt[t * BCDT_W + nt * 16 + laneM] = acc[v];
    }
}

// ---------------- Kernel 2: delta_sp = softplus(dt_low @ W_dtp + b_dt) ----------------
__global__ void __launch_bounds__(32)
gemm_delta(const float* __restrict__ bcdt, const float* __restrict__ Wdtp,
           const float* __restrict__ b_dt, float* __restrict__ delta) {
    const int NT = DDIM / 16;                     // 48
    const int nt = blockIdx.x % NT;
    const int mt = blockIdx.x / NT;
    const int laneM  = threadIdx.x & 15;
    const int laneHi = threadIdx.x >> 4;
    const int m0 = mt * 16, n0 = nt * 16;
    v8f acc = {};
    for (int k0 = 0; k0 < RR; k0 += 4) {
        const int ka = k0 + 2 * laneHi;
        v2f a, b;
        a.x = bcdt[(m0 + laneM) * BCDT_W + 32 + ka];
        a.y = bcdt[(m0 + laneM) * BCDT_W + 32 + ka + 1];
        b.x = Wdtp[(ka    ) * DDIM + n0 + laneM];
        b.y = Wdtp[(ka + 1) * DDIM + n0 + laneM];
        acc = __builtin_amdgcn_wmma_f32_16x16x4_f32(false, a, false, b,
                                                    (short)0, acc, false, false);
    }
    const float bias = b_dt[n0 + laneM];
#pragma unroll
    for (int v = 0; v < 8; ++v) {
        const int t = m0 + v + 8 * laneHi;
        delta[t * DDIM + n0 + laneM] = softplus_f(acc[v] + bias);
    }
}

// ---------------- Kernel 3: per-row RMS norm + soft clamp ----------------
__global__ void __launch_bounds__(256)
rms_clamp(float* __restrict__ delta, const float* __restrict__ rms_scale) {
    __shared__ float red[256];
    const int t = blockIdx.x;
    float v0[3];
    float ss = 0.f;
#pragma unroll
    for (int i = 0; i < 3; ++i) {
        const int d = threadIdx.x + i * 256;
        const float v = delta[t * DDIM + d];
        v0[i] = v;
        ss += v * v;
    }
    red[threadIdx.x] = ss;
    __syncthreads();
    for (int s = 128; s > 0; s >>= 1) {
        if (threadIdx.x < s) red[threadIdx.x] += red[threadIdx.x + s];
        __syncthreads();
    }
    const float inv = rsqrtf(red[0] / (float)DDIM + 1e-6f);
#pragma unroll
    for (int i = 0; i < 3; ++i) {
        const int d = threadIdx.x + i * 256;
        delta[t * DDIM + d] = soft_clamp_f(v0[i] * inv * rms_scale[d]);
    }
}

// ---------------- Recurrence coefficients (bilinear discretization) ----------------
__device__ __forceinline__ void step_coef(const float* __restrict__ x,
                                          const float* __restrict__ delta,
                                          const float* __restrict__ bcdt,
                                          int t, int d, int n, float A,
                                          float& abar, float& bx) {
    const float dE = delta[t * DDIM + d];
    const float hf = 0.5f * dE * A;
    const float denom = 1.0f - hf;
    abar = (1.0f + hf) / denom;
    const float xp  = (t > 0) ? x[(t - 1) * DDIM + d] : 0.0f;
    const float xin = 0.5f * (x[t * DDIM + d] + xp);
    const float Bv  = bcdt[t * BCDT_W + n];
    bx = sqrtf(dE) * Bv / denom * xin;
}

// ---------------- Kernel 4A: per-chunk (prod A_bar, h_end) ----------------
__global__ void __launch_bounds__(256)
scan_passA(const float* __restrict__ x, const float* __restrict__ delta,
           const float* __restrict__ bcdt, const float* __restrict__ A_log,
           float* __restrict__ prodA, float* __restrict__ hend) {
    const int DB    = DDIM / 16;
    const int chunk = blockIdx.x / DB;
    const int dblk  = blockIdx.x % DB;
    const int n = threadIdx.x & 15;
    const int d = dblk * 16 + (threadIdx.x >> 4);
    const float A = -expf(A_log[d * NS + n]);
    float h = 0.f, pA = 1.f;
    const int t0 = chunk * CHLEN;
    for (int t = t0; t < t0 + CHLEN; ++t) {
        float abar, bx;
        step_coef(x, delta, bcdt, t, d, n, A, abar, bx);
        h  = abar * h + bx;
        pA *= abar;
    }
    const int idx = chunk * (DDIM * NS) + d * NS + n;
    prodA[idx] = pA;
    hend[idx]  = h;
}

// ---------------- Kernel 4B: scan chunk summaries; hinit overwrites prodA ----------------
__global__ void __launch_bounds__(256)
scan_passB(float* __restrict__ prodA, const float* __restrict__ hend) {
    const int idx = blockIdx.x * blockDim.x + threadIdx.x;   // d*NS + n
    float h = 0.f;
    for (int c = 0; c < NCH; ++c) {
        const int o = c * (DDIM * NS) + idx;
        const float pa = prodA[o];
        const float he = hend[o];
        prodA[o] = h;                 // h_init for this chunk
        h = pa * h + he;
    }
}

// ---------------- Kernel 4C: replay chunk, reduce over N, write y ----------------
__global__ void __launch_bounds__(256)
scan_passC(const float* __restrict__ x, const float* __restrict__ delta,
           const float* __restrict__ bcdt, const float* __restrict__ A_log,
           const float* __restrict__ hinit, const float* __restrict__ Dp,
           float* __restrict__ y) {
    const int DB    = DDIM / 16;
    const int chunk = blockIdx.x / DB;
    const int dblk  = blockIdx.x % DB;
    const int n = threadIdx.x & 15;
    const int d = dblk * 16 + (threadIdx.x >> 4);
    const float A = -expf(A_log[d * NS + n]);
    float h = hinit[chunk * (DDIM * NS) + d * NS + n];
    const int t0 = chunk * CHLEN;
    for (int t = t0; t < t0 + CHLEN; ++t) {
        float abar, bx;
        step_coef(x, delta, bcdt, t, d, n, A, abar, bx);
        h = abar * h + bx;
        float contrib = bcdt[t * BCDT_W + 16 + n] * h;   // C_sel * h
        // reduce over n = 16 lanes (masks < 16 keep traffic inside each 16-lane group)
        for (int m = 1; m < 16; m <<= 1)
            contrib += __shfl_xor(contrib, m, 32);
        if (n == 0)
            y[t * DDIM + d] = contrib + Dp[d] * x[t * DDIM + d];
    }
}

extern "C" void kernel_launch(void* const* d_in, const int* in_sizes, int n_in,
                              void* d_out, int out_size, void* d_ws, size_t ws_size,
                              hipStream_t stream) {
    const float* x     = (const float*)d_in[0];
    const float* A_log = (const float*)d_in[1];
    const float* Dp    = (const float*)d_in[2];
    const float* Wbc   = (const float*)d_in[3];
    const float* Wdt   = (const float*)d_in[4];
    const float* Wdtp  = (const float*)d_in[5];
    const float* b_dt  = (const float*)d_in[6];
    const float* rs    = (const float*)d_in[7];
    float* y  = (float*)d_out;
    float* ws = (float*)d_ws;

    float* bcdt  = ws;                                   // L*80
    float* delta = bcdt  + (size_t)LSEQ * BCDT_W;        // L*D
    float* prodA = delta + (size_t)LSEQ * DDIM;          // NCH*D*NS (becomes h_init)
    float* hend  = prodA + (size_t)NCH * DDIM * NS;      // NCH*D*NS

    gemm_bcdt <<<dim3((LSEQ / 16) * 5),           dim3(32),  0, stream>>>(x, Wbc, Wdt, bcdt);
    gemm_delta<<<dim3((LSEQ / 16) * (DDIM / 16)), dim3(32),  0, stream>>>(bcdt, Wdtp, b_dt, delta);
    rms_clamp <<<dim3(LSEQ),                      dim3(256), 0, stream>>>(delta, rs);
    scan_passA<<<dim3(NCH * (DDIM / 16)),         dim3(256), 0, stream>>>(x, delta, bcdt, A_log, prodA, hend);
    scan_passB<<<dim3((DDIM * NS) / 256),         dim3(256), 0, stream>>>(prodA, hend);
    scan_passC<<<dim3(NCH * (DDIM / 16)),         dim3(256), 0, stream>>>(x, delta, bcdt, A_log, prodA, Dp, y);
}